// GCN_73332271612655
// MI455X (gfx1250) — compile-verified
//
#include <hip/hip_runtime.h>
#include <hip/hip_bf16.h>

#define NF 256   // input features
#define NH 64    // hidden features
#define BN_EPS 1e-5f
#define KEEP_P 0.7f

typedef float v2f __attribute__((ext_vector_type(2)));
typedef float v8f __attribute__((ext_vector_type(8)));

// ---------------------------------------------------------------------------
// Kernel 1: support = x @ W via V_WMMA_F32_16X16X4_F32.
// Block = 128 threads (4 waves). Each wave owns a 16-row tile and all 64 cols
// (4 n-tiles of 16). W (256x64 fp32 = 64KB) staged in LDS once per block.
// A-frag (16x4 f32, 2 VGPR): lanes 0-15 hold K=k,k+1; lanes 16-31 K=k+2,k+3.
// B-frag (4x16 f32, 2 VGPR): same K split, N = lane%16.
// C/D (16x16 f32, 8 VGPR): VGPR j, lane L -> M = j + 8*(L>=16), N = L%16.
// ---------------------------------------------------------------------------
__global__ __launch_bounds__(128)
void gcn_gemm_wmma(const float* __restrict__ x, const float* __restrict__ W,
                   float* __restrict__ support, int N)
{
    __shared__ float sW[NF * NH];   // 65536 bytes
    for (int i = threadIdx.x; i < NF * NH; i += blockDim.x)
        sW[i] = W[i];
    __syncthreads();

    const int wave = threadIdx.x >> 5;    // 0..3
    const int lane = threadIdx.x & 31;
    const int half = lane >> 4;           // 0 or 1 (K sub-pair)
    const int l16  = lane & 15;
    const int arow = blockIdx.x * 64 + wave * 16 + l16;
    const bool inb = arow < N;
    const float* xrow = x + (size_t)(inb ? arow : (N - 1)) * NF;

    v8f c0 = {}, c1 = {}, c2 = {}, c3 = {};

    #pragma unroll 4
    for (int k = 0; k < NF; k += 4) {
        v2f a;
        float ax = xrow[k + 2 * half + 0];
        float ay = xrow[k + 2 * half + 1];
        a.x = inb ? ax : 0.0f;
        a.y = inb ? ay : 0.0f;

        const float* wk0 = &sW[(k + 2 * half) * NH];
        const float* wk1 = wk0 + NH;
        v2f b0; b0.x = wk0[ 0 + l16]; b0.y = wk1[ 0 + l16];
        v2f b1; b1.x = wk0[16 + l16]; b1.y = wk1[16 + l16];
        v2f b2; b2.x = wk0[32 + l16]; b2.y = wk1[32 + l16];
        v2f b3; b3.x = wk0[48 + l16]; b3.y = wk1[48 + l16];

        c0 = __builtin_amdgcn_wmma_f32_16x16x4_f32(false, a, false, b0, (short)0, c0, false, false);
        c1 = __builtin_amdgcn_wmma_f32_16x16x4_f32(false, a, false, b1, (short)0, c1, false, false);
        c2 = __builtin_amdgcn_wmma_f32_16x16x4_f32(false, a, false, b2, (short)0, c2, false, false);
        c3 = __builtin_amdgcn_wmma_f32_16x16x4_f32(false, a, false, b3, (short)0, c3, false, false);
    }

    const int mbase = blockIdx.x * 64 + wave * 16 + 8 * half;
    #pragma unroll
    for (int j = 0; j < 8; ++j) {
        int r = mbase + j;
        if (r < N) {
            float* o = support + (size_t)r * NH + l16;
            o[ 0] = c0[j];
            o[16] = c1[j];
            o[32] = c2[j];
            o[48] = c3[j];
        }
    }
}

// ---------------------------------------------------------------------------
// Kernel 2: agg[n][f] = b[f]; block 0 also zeroes the stats scratch.
// ---------------------------------------------------------------------------
__global__ __launch_bounds__(256)
void gcn_init(float* __restrict__ agg, const float* __restrict__ b,
              float* __restrict__ stats, long long total)
{
    if (blockIdx.x == 0) stats[threadIdx.x] = 0.0f;
    long long i = (long long)blockIdx.x * 256 + threadIdx.x;
    if (i < total) agg[i] = b[(int)(i & (NH - 1))];
}

// ---------------------------------------------------------------------------
// Kernel 3: COO scatter-add: agg[row[e]] += ew[e] * support[col[e]].
// One thread per (edge, feature-pair); support/agg are L2-resident (51 MB
// combined vs 192 MB L2), so gathers + fp32 atomics stay on-chip.
// ---------------------------------------------------------------------------
__global__ __launch_bounds__(256)
void gcn_edges(const float* __restrict__ support, const int* __restrict__ row,
               const int* __restrict__ col, const float* __restrict__ ew,
               float* __restrict__ agg, long long E)
{
    long long i = (long long)blockIdx.x * 256 + threadIdx.x;
    long long work = E * 32;                // 32 float2 slots per edge
    if (i >= work) return;
    long long e = i >> 5;
    int fp = (int)(i & 31) * 2;
    int c = col[e];
    int r = row[e];
    float w = ew[e];
    const float2 s = *(const float2*)(support + (size_t)c * NH + fp);
    float* dst = agg + (size_t)r * NH + fp;
    atomicAdd(dst + 0, w * s.x);
    atomicAdd(dst + 1, w * s.y);
}

// ---------------------------------------------------------------------------
// Kernel 4: per-column sum / sum-of-squares over rows (LDS tree + atomics).
// Thread (g,f): g = tid/64 walks rows with stride 4 in a 256-row block chunk,
// f = tid%64 is the column -> fully coalesced reads.
// ---------------------------------------------------------------------------
__global__ __launch_bounds__(256)
void gcn_stats(const float* __restrict__ agg, float* __restrict__ stats, int N)
{
    __shared__ float sSum[4][NH];
    __shared__ float sSq[4][NH];
    const int f = threadIdx.x & (NH - 1);
    const int g = threadIdx.x >> 6;
    const int rbase = blockIdx.x * 256;
    const int rend = (rbase + 256 < N) ? (rbase + 256) : N;
    float s = 0.0f, q = 0.0f;
    for (int r = rbase + g; r < rend; r += 4) {
        float v = agg[(size_t)r * NH + f];
        s += v;
        q += v * v;
    }
    sSum[g][f] = s;
    sSq[g][f]  = q;
    __syncthreads();
    if (threadIdx.x < NH) {
        float ts = sSum[0][f] + sSum[1][f] + sSum[2][f] + sSum[3][f];
        float tq = sSq[0][f]  + sSq[1][f]  + sSq[2][f]  + sSq[3][f];
        atomicAdd(&stats[f], ts);
        atomicAdd(&stats[NH + f], tq);
    }
}

// ---------------------------------------------------------------------------
// Kernel 5: fold stats into per-column affine: scale = gamma*rsqrt(var+eps),
// shift = beta - mean*scale.
// ---------------------------------------------------------------------------
__global__ void gcn_finalize(float* __restrict__ stats,
                             const float* __restrict__ gamma,
                             const float* __restrict__ beta, int N)
{
    int f = threadIdx.x;
    if (f < NH) {
        float invN  = 1.0f / (float)N;
        float mean  = stats[f] * invN;
        float var   = stats[NH + f] * invN - mean * mean;
        float inv   = rsqrtf(var + BN_EPS);
        float scale = gamma[f] * inv;
        stats[128 + f] = scale;
        stats[192 + f] = beta[f] - mean * scale;
    }
}

// ---------------------------------------------------------------------------
// threefry2x32-20 (key = (0,42), counter = flat index) for dropout mask.
// ---------------------------------------------------------------------------
__device__ __forceinline__ unsigned tf_rotl(unsigned x, unsigned r)
{
    return (x << r) | (x >> (32u - r));
}

__device__ __forceinline__ unsigned threefry2x32_x0(unsigned k0, unsigned k1,
                                                    unsigned c0, unsigned c1)
{
    unsigned ks[3] = { k0, k1, k0 ^ k1 ^ 0x1BD11BDAu };
    unsigned x0 = c0 + k0, x1 = c1 + k1;
    const unsigned R0[4] = {13u, 15u, 26u, 6u};
    const unsigned R1[4] = {17u, 29u, 16u, 24u};
    #pragma unroll
    for (int blk = 0; blk < 5; ++blk) {
        #pragma unroll
        for (int r = 0; r < 4; ++r) {
            unsigned rot = (blk & 1) ? R1[r] : R0[r];
            x0 += x1;
            x1 = tf_rotl(x1, rot);
            x1 ^= x0;
        }
        x0 += ks[(blk + 1) % 3];
        x1 += ks[(blk + 2) % 3] + (unsigned)(blk + 1);
    }
    return x0;
}

// ---------------------------------------------------------------------------
// Kernel 6: BN affine + ReLU + dropout (inverted scaling by 1/keep_p).
// ---------------------------------------------------------------------------
__global__ __launch_bounds__(256)
void gcn_bn_relu_dropout(const float* __restrict__ agg,
                         const float* __restrict__ stats,
                         float* __restrict__ out, long long total)
{
    long long i = (long long)blockIdx.x * 256 + threadIdx.x;
    if (i >= total) return;
    int f = (int)(i & (NH - 1));
    float scale = stats[128 + f];
    float shift = stats[192 + f];
    float v = fmaf(agg[i], scale, shift);
    v = fmaxf(v, 0.0f);
    unsigned bits = threefry2x32_x0(0u, 42u, (unsigned)(i >> 32), (unsigned)i);
    float u = (float)(bits >> 8) * (1.0f / 16777216.0f);
    out[i] = (u < KEEP_P) ? v * (1.0f / KEEP_P) : 0.0f;
}

// ---------------------------------------------------------------------------
// Host launcher. Inputs: x, row, col, edge_weight, W, b, gamma, beta.
// ws layout: support[N*64] | agg[N*64] | stats[256].
// ---------------------------------------------------------------------------
extern "C" void kernel_launch(void* const* d_in, const int* in_sizes, int n_in,
                              void* d_out, int out_size, void* d_ws, size_t ws_size,
                              hipStream_t stream)
{
    const float* x     = (const float*)d_in[0];
    const int*   row   = (const int*)  d_in[1];
    const int*   col   = (const int*)  d_in[2];
    const float* ew    = (const float*)d_in[3];
    const float* W     = (const float*)d_in[4];
    const float* b     = (const float*)d_in[5];
    const float* gamma = (const float*)d_in[6];
    const float* beta  = (const float*)d_in[7];

    const int       N = in_sizes[0] / NF;
    const long long E = in_sizes[1];
    const long long total = (long long)N * NH;

    float* support = (float*)d_ws;
    float* agg     = support + (size_t)N * NH;
    float* stats   = agg + (size_t)N * NH;
    float* out     = (float*)d_out;

    const int elemBlocks = (int)((total + 255) / 256);

    gcn_gemm_wmma<<<(N + 63) / 64, 128, 0, stream>>>(x, W, support, N);
    gcn_init<<<elemBlocks, 256, 0, stream>>>(agg, b, stats, total);
    {
        long long work = E * 32;
        int blocks = (int)((work + 255) / 256);
        gcn_edges<<<blocks, 256, 0, stream>>>(support, row, col, ew, agg, E);
    }
    gcn_stats<<<(N + 255) / 256, 256, 0, stream>>>(agg, stats, N);
    gcn_finalize<<<1, 64, 0, stream>>>(stats, gamma, beta, N);
    gcn_bn_relu_dropout<<<elemBlocks, 256, 0, stream>>>(agg, stats, out, total);
}